// DotLayer_49289044689062
// MI455X (gfx1250) — compile-verified
//
#include <hip/hip_runtime.h>
#include <math.h>

// -----------------------------------------------------------------------------
// Per-row cosine similarity: out[r] = <a_r,b_r> / (||a_r|| * ||b_r||)
// a,b: [32,2048,1024] f32.  Pure HBM-bandwidth-bound (537 MB stream, ~23 us
// floor at 23.3 TB/s, only ~0.4 GFLOP => 0.75 FLOP/B; WMMA irrelevant).
// CDNA5 path used: ASYNCcnt global->LDS copy pipeline (4-deep per-wave
// double-buffering, zero VGPR load-destination pressure) + wave32 shuffles.
// -----------------------------------------------------------------------------

#ifndef __has_builtin
#define __has_builtin(x) 0
#endif

#define ROWS_PER_BLOCK 8
#define ROW_D 1024
#define CHUNK 128                // floats per chunk per operand = 1 b128/lane
#define NCHUNK (ROW_D / CHUNK)   // 8
#define PIPE 4                   // async pipeline depth (chunks in flight)

// Device-pass-only feature detection (host pass must not see amdgcn builtins).
#if defined(__HIP_DEVICE_COMPILE__) && \
    __has_builtin(__builtin_amdgcn_global_load_async_to_lds_b128)
#define ASYNC_PATH 1
#else
#define ASYNC_PATH 0
#endif

#if ASYNC_PATH
#define AS1 __attribute__((address_space(1)))
#define AS3 __attribute__((address_space(3)))
// Builtin params are typed pointers: v4i in AS1 (global) and v4i in AS3 (LDS).
typedef int v4i __attribute__((vector_size(16)));
#if __has_builtin(__builtin_amdgcn_s_wait_asynccnt)
#define WAIT_ASYNC(n) __builtin_amdgcn_s_wait_asynccnt(n)
#else
#define WAIT_ASYNC(n) asm volatile("s_wait_asynccnt " #n ::: "memory")
#endif
#endif

__device__ __forceinline__ float wave32_reduce_add(float v) {
  // wave32 butterfly (CDNA5 is wave32-only; do NOT assume 64 lanes).
#pragma unroll
  for (int m = 16; m > 0; m >>= 1) v += __shfl_xor(v, m, 32);
  return v;
}

__global__ __launch_bounds__(256)
void cosine_rows_kernel(const float* __restrict__ a,
                        const float* __restrict__ b,
                        float* __restrict__ out, int nrows) {
  const int lane = threadIdx.x & 31;
  const int wave = threadIdx.x >> 5;
  const int row  = blockIdx.x * ROWS_PER_BLOCK + wave;
  if (row >= nrows) return;

  const float* __restrict__ ga = a + (size_t)row * ROW_D;
  const float* __restrict__ gb = b + (size_t)row * ROW_D;

  float dot = 0.f, saa = 0.f, sbb = 0.f;

#if ASYNC_PATH
  // Per-wave private ring buffers: 8 waves * 4 bufs * 128 f32 * 2 ops = 32KB
  // (vs 320KB/WGP LDS pool -> no occupancy impact). 8 async b128 ops (4KB)
  // in flight per wave for HBM latency tolerance.
  __shared__ __align__(16) float ldsA[ROWS_PER_BLOCK][PIPE][CHUNK];
  __shared__ __align__(16) float ldsB[ROWS_PER_BLOCK][PIPE][CHUNK];

  const int fo = lane * 4;  // 16B per lane -> 512B coalesced per wave issue

  auto issue = [&](int c) {
    const int p = c & (PIPE - 1);
    __builtin_amdgcn_global_load_async_to_lds_b128(
        (AS1 v4i*)(ga + c * CHUNK + fo), (AS3 v4i*)(&ldsA[wave][p][fo]),
        /*offset=*/0, /*cpol=*/0);
    __builtin_amdgcn_global_load_async_to_lds_b128(
        (AS1 v4i*)(gb + c * CHUNK + fo), (AS3 v4i*)(&ldsB[wave][p][fo]),
        /*offset=*/0, /*cpol=*/0);
  };

#pragma unroll
  for (int c = 0; c < PIPE; ++c) issue(c);

#pragma unroll
  for (int c = 0; c < NCHUNK; ++c) {
    // Async loads complete in order. Before computing chunk c, at most
    // NCHUNK-1-c later chunks (2 ops each) may remain outstanding, capped by
    // pipeline depth. s_wait_asynccnt needs an immediate -> constant ladder.
    const int rem = NCHUNK - 1 - c;  // chunks allowed to stay in flight
    if (rem >= PIPE - 1) {
      WAIT_ASYNC(6);
    } else if (rem == 2) {
      WAIT_ASYNC(4);
    } else if (rem == 1) {
      WAIT_ASYNC(2);
    } else {
      WAIT_ASYNC(0);
    }
    asm volatile("" ::: "memory");  // keep LDS reads below the wait

    const int p = c & (PIPE - 1);
    const float4 av = *reinterpret_cast<const float4*>(&ldsA[wave][p][fo]);
    const float4 bv = *reinterpret_cast<const float4*>(&ldsB[wave][p][fo]);

    dot = fmaf(av.x, bv.x, dot); dot = fmaf(av.y, bv.y, dot);
    dot = fmaf(av.z, bv.z, dot); dot = fmaf(av.w, bv.w, dot);
    saa = fmaf(av.x, av.x, saa); saa = fmaf(av.y, av.y, saa);
    saa = fmaf(av.z, av.z, saa); saa = fmaf(av.w, av.w, saa);
    sbb = fmaf(bv.x, bv.x, sbb); sbb = fmaf(bv.y, bv.y, sbb);
    sbb = fmaf(bv.z, bv.z, sbb); sbb = fmaf(bv.w, bv.w, sbb);

    if (c + PIPE < NCHUNK) issue(c + PIPE);
  }
#else
  // Fallback: direct b128 streaming loads (also bandwidth-optimal).
#pragma unroll
  for (int c = 0; c < NCHUNK; ++c) {
    const float4 av =
        *reinterpret_cast<const float4*>(ga + c * CHUNK + lane * 4);
    const float4 bv =
        *reinterpret_cast<const float4*>(gb + c * CHUNK + lane * 4);
    dot = fmaf(av.x, bv.x, dot); dot = fmaf(av.y, bv.y, dot);
    dot = fmaf(av.z, bv.z, dot); dot = fmaf(av.w, bv.w, dot);
    saa = fmaf(av.x, av.x, saa); saa = fmaf(av.y, av.y, saa);
    saa = fmaf(av.z, av.z, saa); saa = fmaf(av.w, av.w, saa);
    sbb = fmaf(bv.x, bv.x, sbb); sbb = fmaf(bv.y, bv.y, sbb);
    sbb = fmaf(bv.z, bv.z, sbb); sbb = fmaf(bv.w, bv.w, sbb);
  }
#endif

  dot = wave32_reduce_add(dot);
  saa = wave32_reduce_add(saa);
  sbb = wave32_reduce_add(sbb);

  if (lane == 0) {
    // matches tf.math.l2_normalize: sqrt(max(sum_sq, 1e-12))
    const float denom =
        sqrtf(fmaxf(saa, 1e-12f)) * sqrtf(fmaxf(sbb, 1e-12f));
    out[row] = dot / denom;
  }
}

extern "C" void kernel_launch(void* const* d_in, const int* in_sizes, int n_in,
                              void* d_out, int out_size, void* d_ws,
                              size_t ws_size, hipStream_t stream) {
  (void)n_in; (void)d_ws; (void)ws_size; (void)in_sizes;
  const float* a = (const float*)d_in[0];
  const float* b = (const float*)d_in[1];
  float* out = (float*)d_out;
  const int nrows = out_size;  // 32*2048 = 65536 rows of depth 1024

  dim3 block(256);  // 8 wave32 waves
  dim3 grid((nrows + ROWS_PER_BLOCK - 1) / ROWS_PER_BLOCK);
  hipLaunchKernelGGL(cosine_rows_kernel, grid, block, 0, stream, a, b, out,
                     nrows);
}